// GCN_53498112639197
// MI455X (gfx1250) — compile-verified
//
#include <hip/hip_runtime.h>
#include <hip/hip_bf16.h>

typedef float v2f __attribute__((ext_vector_type(2)));
typedef float v8f __attribute__((ext_vector_type(8)));

#define N_NODES   100000
#define N_EDGES   1600000
#define IN_C      256
#define HID_C     128
#define NUM_GRAPHS 64
#define BN_EPS    1e-5f
#define NF        128   // feature width of every intermediate (HID_C == OUT_C == 128)

// ---------------------------------------------------------------- degree / norm
__global__ void k_init_deg(float* __restrict__ deg) {
    int i = blockIdx.x * blockDim.x + threadIdx.x;
    if (i < N_NODES) deg[i] = 1.0f;              // self-loop included
}

__global__ void k_deg_accum(const int* __restrict__ col, float* __restrict__ deg) {
    int e = blockIdx.x * blockDim.x + threadIdx.x;
    if (e < N_EDGES) atomicAdd(&deg[col[e]], 1.0f);
}

__global__ void k_dis(const float* __restrict__ deg, float* __restrict__ dis) {
    int i = blockIdx.x * blockDim.x + threadIdx.x;
    if (i < N_NODES) dis[i] = rsqrtf(deg[i]);
}

// ---------------------------------------------------------------- WMMA GEMM
// C[M,128] = A[M,K] @ B[K,128].  One wave computes a 16-row x 128-col strip:
// A fragment (float2 per lane) is loaded once per K-step and reused across the
// 8 column tiles (8 v8f accumulators).  Uses V_WMMA_F32_16X16X4_F32.
__global__ __launch_bounds__(256)
void k_wmma_gemm_rowstrip(const float* __restrict__ A, const float* __restrict__ B,
                          float* __restrict__ C, int M, int K) {
    int wid  = (blockIdx.x * blockDim.x + threadIdx.x) >> 5;
    int lane = threadIdx.x & 31;
    int mtiles = M >> 4;
    if (wid >= mtiles) return;

    long row0 = (long)wid * 16;
    int  m    = lane & 15;           // M index (A) / N index (B, store)
    int  kb   = (lane >> 4) << 1;    // lanes 16-31 hold K+2,K+3

    v8f acc[8];
#pragma unroll
    for (int t = 0; t < 8; ++t)
#pragma unroll
        for (int j = 0; j < 8; ++j) acc[t][j] = 0.0f;

    const float* arow = A + (row0 + m) * (long)K;
    for (int k0 = 0; k0 < K; k0 += 4) {
        v2f a = *(const v2f*)(arow + k0 + kb);
        const float* bp0 = B + (long)(k0 + kb) * NF;   // row k0+kb
        const float* bp1 = bp0 + NF;                   // row k0+kb+1
#pragma unroll
        for (int t = 0; t < 8; ++t) {
            v2f b;
            b.x = bp0[t * 16 + m];
            b.y = bp1[t * 16 + m];
            acc[t] = __builtin_amdgcn_wmma_f32_16x16x4_f32(
                false, a, false, b, (short)0, acc[t], false, false);
        }
    }

    int mb = (lane >> 4) * 8;        // C/D: lanes 16-31 hold rows M=8..15
#pragma unroll
    for (int t = 0; t < 8; ++t)
#pragma unroll
        for (int j = 0; j < 8; ++j)
            C[(row0 + mb + j) * NF + t * 16 + m] = acc[t][j];
}

// ---------------------------------------------------------------- self-loop init
// agg[i,f] = dis[i]^2 * xw[i,f] + bias[f]
__global__ void k_agg_init(const float* __restrict__ xw, const float* __restrict__ dis,
                           const float* __restrict__ bias, float* __restrict__ agg) {
    long i = (long)blockIdx.x * blockDim.x + threadIdx.x;
    if (i >= (long)N_NODES * NF) return;
    int node = (int)(i >> 7);
    int f    = (int)(i & 127);
    float d = dis[node];
    agg[i] = d * d * xw[i] + bias[f];
}

// ---------------------------------------------------------------- edge scatter
// one wave per edge: 32 lanes x float4 = 128 features gathered from xw[row],
// scaled by dis[row]*dis[col], atomically added into agg[col].
__global__ __launch_bounds__(256)
void k_scatter(const float* __restrict__ xw, const float* __restrict__ dis,
               const int* __restrict__ row, const int* __restrict__ col,
               float* __restrict__ agg) {
    long gid  = (long)blockIdx.x * blockDim.x + threadIdx.x;
    int  e    = (int)(gid >> 5);
    int  lane = threadIdx.x & 31;
    if (e >= N_EDGES) return;
    int r = row[e], c = col[e];
    float norm = dis[r] * dis[c];
    const float4* src = (const float4*)(xw + (long)r * NF);
    float4 v = src[lane];
    float* dst = agg + (long)c * NF + lane * 4;
    atomicAdd(dst + 0, norm * v.x);
    atomicAdd(dst + 1, norm * v.y);
    atomicAdd(dst + 2, norm * v.z);
    atomicAdd(dst + 3, norm * v.w);
}

// ---------------------------------------------------------------- batchnorm
__global__ void k_zero_stats(float* __restrict__ sum, float* __restrict__ sumsq) {
    int i = threadIdx.x;
    if (i < NF) { sum[i] = 0.0f; sumsq[i] = 0.0f; }
}

__global__ void k_bn_stats(const float* __restrict__ x, float* __restrict__ sum,
                           float* __restrict__ sumsq) {
    int f  = threadIdx.x;                  // 0..127, one feature per thread
    int r0 = blockIdx.x * 512;
    float s = 0.0f, s2 = 0.0f;
    for (int i = 0; i < 512; ++i) {
        int r = r0 + i;
        if (r >= N_NODES) break;
        float v = x[(long)r * NF + f];
        s += v; s2 += v * v;
    }
    atomicAdd(&sum[f], s);
    atomicAdd(&sumsq[f], s2);
}

__global__ void k_bn_final(const float* __restrict__ sum, const float* __restrict__ sumsq,
                           float* __restrict__ mean, float* __restrict__ invstd) {
    int f = threadIdx.x;
    if (f >= NF) return;
    float mu  = sum[f] / (float)N_NODES;
    float var = sumsq[f] / (float)N_NODES - mu * mu;   // biased, training-mode
    mean[f]   = mu;
    invstd[f] = rsqrtf(var + BN_EPS);
}

__global__ void k_bn_apply_relu(const float* __restrict__ x, const float* __restrict__ gamma,
                                const float* __restrict__ beta, const float* __restrict__ mean,
                                const float* __restrict__ invstd, float* __restrict__ y) {
    long i = (long)blockIdx.x * blockDim.x + threadIdx.x;
    if (i >= (long)N_NODES * NF) return;
    int f = (int)(i & 127);
    float v = gamma[f] * (x[i] - mean[f]) * invstd[f] + beta[f];
    y[i] = fmaxf(v, 0.0f);
}

// ---------------------------------------------------------------- pooling
__global__ void k_zero_pool(float* __restrict__ psum, float* __restrict__ cnt) {
    int i = blockIdx.x * blockDim.x + threadIdx.x;
    if (i < NUM_GRAPHS * NF) psum[i] = 0.0f;
    if (i < NUM_GRAPHS)      cnt[i]  = 0.0f;
}

__global__ void k_pool_accum(const float* __restrict__ h, const int* __restrict__ batch,
                             float* __restrict__ psum, float* __restrict__ cnt) {
    long i = (long)blockIdx.x * blockDim.x + threadIdx.x;
    if (i >= (long)N_NODES * NF) return;
    int node = (int)(i >> 7);
    int f    = (int)(i & 127);
    int g    = batch[node];
    atomicAdd(&psum[g * NF + f], h[i]);
    if (f == 0) atomicAdd(&cnt[g], 1.0f);
}

__global__ void k_pool_div(const float* __restrict__ psum, const float* __restrict__ cnt,
                           float* __restrict__ pooled) {
    int i = blockIdx.x * blockDim.x + threadIdx.x;
    if (i >= NUM_GRAPHS * NF) return;
    int g = i >> 7;
    pooled[i] = psum[i] / fmaxf(cnt[g], 1.0f);
}

// ---------------------------------------------------------------- final GEMM
// out[64,128] = pooled[64,128] @ linW^T + linb;   B(k,n) = linW[n*128+k]
__global__ __launch_bounds__(256)
void k_wmma_gemm_bt_bias(const float* __restrict__ A, const float* __restrict__ Wt,
                         const float* __restrict__ bias, float* __restrict__ C) {
    int wid  = (blockIdx.x * blockDim.x + threadIdx.x) >> 5;
    int lane = threadIdx.x & 31;
    if (wid >= 32) return;               // 4 M-tiles x 8 N-tiles
    int tm = wid >> 3, tn = wid & 7;
    int row0 = tm * 16, col0 = tn * 16;
    int m  = lane & 15;
    int kb = (lane >> 4) << 1;

    v8f c;
    float bv = bias[col0 + m];           // depends only on N = lane&15
#pragma unroll
    for (int j = 0; j < 8; ++j) c[j] = bv;

    for (int k0 = 0; k0 < 128; k0 += 4) {
        v2f a = *(const v2f*)(A + (row0 + m) * NF + k0 + kb);
        v2f b = *(const v2f*)(Wt + (col0 + m) * NF + k0 + kb);  // linW row n, contiguous k
        c = __builtin_amdgcn_wmma_f32_16x16x4_f32(
            false, a, false, b, (short)0, c, false, false);
    }

    int mb = (lane >> 4) * 8;
#pragma unroll
    for (int j = 0; j < 8; ++j)
        C[(row0 + mb + j) * NF + col0 + m] = c[j];
}

// ================================================================ launcher
extern "C" void kernel_launch(void* const* d_in, const int* in_sizes, int n_in,
                              void* d_out, int out_size, void* d_ws, size_t ws_size,
                              hipStream_t stream) {
    const float* x      = (const float*)d_in[0];
    const float* W1     = (const float*)d_in[1];
    const float* b1     = (const float*)d_in[2];
    const float* gamma1 = (const float*)d_in[3];
    const float* beta1  = (const float*)d_in[4];
    const float* W2     = (const float*)d_in[5];
    const float* b2     = (const float*)d_in[6];
    const float* gamma2 = (const float*)d_in[7];
    const float* beta2  = (const float*)d_in[8];
    const float* linW   = (const float*)d_in[9];
    const float* linb   = (const float*)d_in[10];
    const int*   eidx   = (const int*)d_in[11];
    const int*   batch  = (const int*)d_in[12];
    float*       out    = (float*)d_out;

    const int* erow = eidx;             // edge_index[0]
    const int* ecol = eidx + N_EDGES;   // edge_index[1]

    // workspace layout (floats)
    float* ws    = (float*)d_ws;
    float* bufA  = ws;                       // 12.8M  (xw1 / h1 / agg2 / ...)
    float* bufB  = bufA + (long)N_NODES*NF;  // 12.8M  (agg1 / xw2 / h2)
    float* deg   = bufB + (long)N_NODES*NF;  // 100000
    float* dis   = deg  + N_NODES;           // 100000
    float* sum   = dis  + N_NODES;           // 128
    float* sumsq = sum  + NF;                // 128
    float* mean  = sumsq+ NF;                // 128
    float* invst = mean + NF;                // 128
    float* psum  = invst+ NF;                // 8192
    float* pooled= psum + NUM_GRAPHS*NF;     // 8192
    float* cnt   = pooled+ NUM_GRAPHS*NF;    // 64

    const int TPB = 256;
    const int nodeBlocks = (N_NODES + TPB - 1) / TPB;
    const int edgeBlocks = (N_EDGES + TPB - 1) / TPB;
    const int nfElems    = N_NODES * NF;             // 12.8M
    const int nfBlocks   = (nfElems + TPB - 1) / TPB;
    const int scatBlocks = (N_EDGES * 32 + TPB - 1) / TPB;   // wave per edge
    const int mtiles     = N_NODES / 16;             // 6250
    const int gemmBlocks = (mtiles + 7) / 8;         // 8 waves per block
    const int bnBlocks   = (N_NODES + 511) / 512;

    // degree + normalization
    k_init_deg <<<nodeBlocks, TPB, 0, stream>>>(deg);
    k_deg_accum<<<edgeBlocks, TPB, 0, stream>>>(ecol, deg);
    k_dis      <<<nodeBlocks, TPB, 0, stream>>>(deg, dis);

    // ---- layer 1 ----
    k_wmma_gemm_rowstrip<<<gemmBlocks, TPB, 0, stream>>>(x, W1, bufA, N_NODES, IN_C);
    k_agg_init <<<nfBlocks,  TPB, 0, stream>>>(bufA, dis, b1, bufB);
    k_scatter  <<<scatBlocks,TPB, 0, stream>>>(bufA, dis, erow, ecol, bufB);
    k_zero_stats<<<1, NF, 0, stream>>>(sum, sumsq);
    k_bn_stats <<<bnBlocks, NF, 0, stream>>>(bufB, sum, sumsq);
    k_bn_final <<<1, NF, 0, stream>>>(sum, sumsq, mean, invst);
    k_bn_apply_relu<<<nfBlocks, TPB, 0, stream>>>(bufB, gamma1, beta1, mean, invst, bufA);

    // ---- layer 2 ----
    k_wmma_gemm_rowstrip<<<gemmBlocks, TPB, 0, stream>>>(bufA, W2, bufB, N_NODES, HID_C);
    k_agg_init <<<nfBlocks,  TPB, 0, stream>>>(bufB, dis, b2, bufA);
    k_scatter  <<<scatBlocks,TPB, 0, stream>>>(bufB, dis, erow, ecol, bufA);
    k_zero_stats<<<1, NF, 0, stream>>>(sum, sumsq);
    k_bn_stats <<<bnBlocks, NF, 0, stream>>>(bufA, sum, sumsq);
    k_bn_final <<<1, NF, 0, stream>>>(sum, sumsq, mean, invst);
    k_bn_apply_relu<<<nfBlocks, TPB, 0, stream>>>(bufA, gamma2, beta2, mean, invst, bufB);

    // ---- pooling + head ----
    k_zero_pool <<<(NUM_GRAPHS*NF + TPB - 1)/TPB, TPB, 0, stream>>>(psum, cnt);
    k_pool_accum<<<nfBlocks, TPB, 0, stream>>>(bufB, batch, psum, cnt);
    k_pool_div  <<<(NUM_GRAPHS*NF + TPB - 1)/TPB, TPB, 0, stream>>>(psum, cnt, pooled);
    k_wmma_gemm_bt_bias<<<4, TPB, 0, stream>>>(pooled, linW, linb, out);

    (void)in_sizes; (void)n_in; (void)out_size; (void)ws_size;
}